// LightGCN_18794776887301
// MI455X (gfx1250) — compile-verified
//
#include <hip/hip_runtime.h>
#include <cstdint>

// ---------------- problem constants (match reference) ----------------
#define N_USERS_C 100000
#define N_ITEMS_C 50000
#define N_TOTAL_C (N_USERS_C + N_ITEMS_C)
#define EMB_C     64
#define LAYERS_C  3

#define TPB 256      // 8 wave32 per block
#define EPB 512      // edges per LDS tile (12 B/edge; 2 buffers -> 12 KB LDS)
#define SPMM_GRID 2048

// ---------------- CDNA5 async global->LDS helpers ----------------
// global_load_async_to_lds_b32: VDST = per-lane LDS byte address (32-bit VGPR),
// VADDR = 64-bit global address (VGPR pair), 'off' = no SADDR. Tracked by ASYNCcnt.
__device__ __forceinline__ void async_copy_b32(uint32_t lds_addr, const void* gaddr) {
    asm volatile("global_load_async_to_lds_b32 %0, %1, off"
                 :: "v"(lds_addr), "v"((unsigned long long)(uintptr_t)gaddr)
                 : "memory");
}

// Wait until at most N of this wave's async copies remain in flight.
// Async loads complete in order, so <=6 means "everything older than the
// newest 6-instruction prefetch batch has landed in LDS".
__device__ __forceinline__ void wait_async_le6() {
    asm volatile("s_wait_asynccnt 0x6" ::: "memory");
}
__device__ __forceinline__ void wait_async_0() {
    asm volatile("s_wait_asynccnt 0x0" ::: "memory");
}

// Low 32 bits of a generic pointer into __shared__ are the LDS byte offset.
__device__ __forceinline__ uint32_t lds_offset(const void* p) {
    return (uint32_t)(uintptr_t)p;
}

// ---------------- kernels ----------------

// x = acc = concat(user_emb, item_emb), float4-vectorized.
__global__ __launch_bounds__(TPB) void lgcn_init(const float4* __restrict__ u,
                                                 const float4* __restrict__ it,
                                                 float4* __restrict__ x,
                                                 float4* __restrict__ acc,
                                                 int n4) {
    int i = blockIdx.x * TPB + threadIdx.x;
    if (i >= n4) return;
    const int u4 = N_USERS_C * (EMB_C / 4);
    float4 v = (i < u4) ? u[i] : it[i - u4];
    x[i] = v;
    acc[i] = v;
}

__global__ __launch_bounds__(TPB) void lgcn_zero(float4* __restrict__ y, int n4) {
    int i = blockIdx.x * TPB + threadIdx.x;
    if (i >= n4) return;
    y[i] = make_float4(0.f, 0.f, 0.f, 0.f);
}

// y[r] += val * x[c] for every edge.
// Grid-stride over 512-edge tiles; double-buffered async global->LDS staging of
// the edge stream overlaps the next tile's HBM fetch with this tile's
// gather/atomic work. One wave32 per edge iteration: lanes cover the 64-dim
// embedding as two contiguous 128 B transactions (loads and atomic adds).
__global__ __launch_bounds__(TPB) void lgcn_spmm(const int* __restrict__ erow,
                                                 const int* __restrict__ ecol,
                                                 const float* __restrict__ eval,
                                                 const float* __restrict__ x,
                                                 float* __restrict__ y,
                                                 int nnz) {
    __shared__ int   s_row[2][EPB];
    __shared__ int   s_col[2][EPB];
    __shared__ float s_val[2][EPB];

    const int tid    = threadIdx.x;
    const int G      = gridDim.x;
    const int ntiles = (nnz + EPB - 1) / EPB;

    // Issue 6 async copy instructions per wave for one tile (2 iters x 3 arrays).
    auto stage = [&](int tile, int buf) {
        const int base = tile * EPB;
        int nloc = nnz - base;
        if (nloc > EPB) nloc = EPB;
        for (int k = tid; k < EPB; k += TPB) {
            int kk = k < nloc ? k : (nloc - 1);   // clamp tail, keep lanes active
            int g  = base + kk;
            async_copy_b32(lds_offset(&s_row[buf][k]), &erow[g]);
            async_copy_b32(lds_offset(&s_col[buf][k]), &ecol[g]);
            async_copy_b32(lds_offset(&s_val[buf][k]), &eval[g]);
        }
    };

    const int t0 = blockIdx.x;
    if (t0 >= ntiles) return;
    stage(t0, 0);                         // prologue prefetch

    const int lane = tid & 31;
    const int wv   = tid >> 5;

    int i = 0;
    for (int tile = t0; tile < ntiles; tile += G, ++i) {
        const int p    = i & 1;
        const int next = tile + G;
        if (next < ntiles) {
            stage(next, p ^ 1);           // prefetch tile t+1 into other buffer
            wait_async_le6();             // tile t's copies (older) have landed
        } else {
            wait_async_0();
        }
        __syncthreads();                  // all waves' tile-t data visible

        const int base = tile * EPB;
        int nloc = nnz - base;
        if (nloc > EPB) nloc = EPB;
        for (int e = wv; e < nloc; e += TPB / 32) {
            const int   r = s_row[p][e];
            const int   c = s_col[p][e];
            const float w = s_val[p][e];
            const size_t cb = (size_t)c * EMB_C;
            const size_t rb = (size_t)r * EMB_C;
            // two coalesced 128 B gathers (L2-resident x)
            float a0 = x[cb + lane];
            float a1 = x[cb + 32 + lane];
            // two coalesced 128 B scatter-adds (global_atomic_add_f32, L2-resident y)
            atomicAdd(&y[rb + lane],      w * a0);
            atomicAdd(&y[rb + 32 + lane], w * a1);
        }
        __syncthreads();                  // done reading buf p before it's re-staged
    }
}

__global__ __launch_bounds__(TPB) void lgcn_accadd(float4* __restrict__ acc,
                                                   const float4* __restrict__ y,
                                                   int n4) {
    int i = blockIdx.x * TPB + threadIdx.x;
    if (i >= n4) return;
    float4 a = acc[i];
    float4 b = y[i];
    a.x += b.x; a.y += b.y; a.z += b.z; a.w += b.w;
    acc[i] = a;
}

__global__ __launch_bounds__(TPB) void lgcn_final(float4* __restrict__ acc, int n4) {
    int i = blockIdx.x * TPB + threadIdx.x;
    if (i >= n4) return;
    float4 a = acc[i];
    const float s = 1.0f / (LAYERS_C + 1);
    a.x *= s; a.y *= s; a.z *= s; a.w *= s;
    acc[i] = a;
}

// ---------------- host-side orchestration ----------------
extern "C" void kernel_launch(void* const* d_in, const int* in_sizes, int n_in,
                              void* d_out, int out_size, void* d_ws, size_t ws_size,
                              hipStream_t stream) {
    const float* uw = (const float*)d_in[0];   // [N_USERS, 64]
    const float* iw = (const float*)d_in[1];   // [N_ITEMS, 64]
    const int*   er = (const int*)d_in[2];     // [NNZ]
    const int*   ec = (const int*)d_in[3];     // [NNZ]
    const float* ev = (const float*)d_in[4];   // [NNZ]
    const int   nnz = in_sizes[2];

    // acc lives directly in d_out ([N_TOTAL, 64] == concatenated output).
    float* acc = (float*)d_out;

    const size_t nelem = (size_t)N_TOTAL_C * EMB_C;   // 9.6M floats
    float* x = (float*)d_ws;                          // ping
    float* y = x + nelem;                             // pong (2 x 38.4 MB total)

    const int n4    = (int)(nelem / 4);
    const int gridE = (n4 + TPB - 1) / TPB;
    const int ntile = (nnz + EPB - 1) / EPB;
    const int gridS = ntile < SPMM_GRID ? ntile : SPMM_GRID;

    lgcn_init<<<gridE, TPB, 0, stream>>>((const float4*)uw, (const float4*)iw,
                                         (float4*)x, (float4*)acc, n4);
    for (int l = 0; l < LAYERS_C; ++l) {
        lgcn_zero<<<gridE, TPB, 0, stream>>>((float4*)y, n4);
        lgcn_spmm<<<gridS, TPB, 0, stream>>>(er, ec, ev, x, y, nnz);
        lgcn_accadd<<<gridE, TPB, 0, stream>>>((float4*)acc, (const float4*)y, n4);
        float* t = x; x = y; y = t;   // next layer reads this layer's output
    }
    lgcn_final<<<gridE, TPB, 0, stream>>>((float4*)acc, n4);
}